// GaussianSplatting3D_80376017977407
// MI455X (gfx1250) — compile-verified
//
#include <hip/hip_runtime.h>
#include <hip/hip_bf16.h>
#include <math.h>

// MI455X (gfx1250) implementation of point-splat gaussian compositing.
//
// Cost model: output zeroing/writing 35*1024*1024 floats = 147MB dominates
// (~6.3us @ 23.3 TB/s). N=8192 splats -> everything else is tiny, so we:
//   k_zero      : float4 grid-stride zero of the output
//   k_project   : WMMA f32 16x16x4 affine projection (16 points / wave),
//                 emits packed {pid, depth, log1p(-o_clip), o} records
//   k_composite : O(N^2) same-pixel transmittance scan over LDS-staged
//                 records (replaces the sort), float atomics for the
//                 rare per-pixel collisions. Records staged with
//                 global_load_async_to_lds_b128 (ASYNCcnt-tracked).

typedef __attribute__((ext_vector_type(2))) float v2f;
typedef __attribute__((ext_vector_type(8))) float v8f;
typedef __attribute__((ext_vector_type(4))) int   v4i;

#define FEAT_DIM 35
#define CHUNK    2048   // records staged to LDS per tile (2048*16B = 32KB)

// ------------------------------------------------------------------
// Kernel 1: zero the 147MB output with wide stores.
// ------------------------------------------------------------------
__global__ void k_zero(float* __restrict__ out, int n) {
  const int n4 = n >> 2;
  float4* __restrict__ o4 = (float4*)out;
  const float4 z = make_float4(0.f, 0.f, 0.f, 0.f);
  int i = blockIdx.x * blockDim.x + threadIdx.x;
  const int stride = gridDim.x * blockDim.x;
  for (; i < n4; i += stride) o4[i] = z;
  const int tail = n & 3;
  if (blockIdx.x == 0 && (int)threadIdx.x < tail) out[(n4 << 2) + threadIdx.x] = 0.f;
}

// ------------------------------------------------------------------
// Kernel 2: project 16 points per wave with one V_WMMA_F32_16X16X4_F32.
//   A (16x4): rows 0..2 = K@[R|t] rows, row 3 = [R|t] row 2 (cam z).
//             16x4 f32 A layout: lanes 0-15 supply K=0,1 in vgpr0,1;
//             lanes 16-31 supply K=2,3.
//   B (4x16): column n = homogeneous point (x,y,z,1) for point base+n.
//   D (16x16): lane n (<16) holds projx,projy,projw,camz in d[0..3].
// ------------------------------------------------------------------
__global__ void k_project(const float* __restrict__ centers,
                          const float* __restrict__ opac,
                          const float* __restrict__ pose,   // 4x4 row-major
                          const float* __restrict__ intr,   // 3x3 row-major
                          const int* __restrict__ Hp, const int* __restrict__ Wp,
                          int N, float4* __restrict__ rec) {
  const int H = *Hp, W = *Wp;
  const int lane = threadIdx.x & 31;
  const int wave = (int)((blockIdx.x * blockDim.x + threadIdx.x) >> 5);
  const int base = wave * 16;
  if (base >= N) return;  // wave-uniform

  const int  r  = lane & 15;
  const bool hi = lane >= 16;

  // ---- A operand (combined camera matrix), built branch-convergent ----
  float a0 = 0.f, a1 = 0.f;
  {
    const int c0 = hi ? 2 : 0;
    const int c1 = c0 + 1;
    if (r < 4) {
      if (r == 3) {               // cam-z row = pose row 2
        a0 = pose[8 + c0];
        a1 = pose[8 + c1];
      } else {                    // (K @ [R|t]) row r
        const float k0 = intr[r * 3 + 0];
        const float k1 = intr[r * 3 + 1];
        const float k2 = intr[r * 3 + 2];
        a0 = k0 * pose[0 + c0] + k1 * pose[4 + c0] + k2 * pose[8 + c0];
        a1 = k0 * pose[0 + c1] + k1 * pose[4 + c1] + k2 * pose[8 + c1];
      }
    }
  }

  // ---- B operand: 16 homogeneous points in columns ----
  const int p  = base + r;
  const int pc = (p < N) ? p : (N - 1);
  float b0, b1;
  if (!hi) { b0 = centers[pc * 3 + 0]; b1 = centers[pc * 3 + 1]; }  // K=0,1 -> x,y
  else     { b0 = centers[pc * 3 + 2]; b1 = 1.0f; }                 // K=2,3 -> z,1

  v2f av; av[0] = a0; av[1] = a1;
  v2f bv; bv[0] = b0; bv[1] = b1;
  v8f cz = {};
  // (neg_a, A, neg_b, B, c_mod, C, reuse_a, reuse_b)
  v8f d = __builtin_amdgcn_wmma_f32_16x16x4_f32(false, av, false, bv,
                                                (short)0, cz, false, false);

  if (!hi && p < N) {
    const float projx = d[0], projy = d[1], projw = d[2], z = d[3];
    const bool  valid = z > 0.1f;
    const float denom = valid ? projw : 1.0f;
    const int   px = (int)(projx / denom);   // trunc-toward-zero == astype(int32)
    const int   py = (int)(projy / denom);
    const bool  ok = valid && (px >= 0) && (px < W) && (py >= 0) && (py < H);
    const int   pid   = ok ? (py * W + px) : (H * W);
    const float o     = ok ? opac[p] : 0.0f;
    const float depth = ok ? z : __builtin_inff();
    const float ocl   = fminf(fmaxf(o, 0.0f), 1.0f - 1e-7f);
    const float logt  = log1pf(-ocl);
    rec[p] = make_float4(__int_as_float(pid), depth, logt, o);
  }
}

// ------------------------------------------------------------------
// Kernel 3: transmittance + scatter. For splat i:
//   tsum = sum of log1p(-o_j) over j with pid_j == pid_i and
//          (depth_j < depth_i or (depth_j == depth_i and j < i))
//   alpha = o_i * exp(tsum);  out[f*HW + pid_i] += alpha * feat_i[f]
// Records staged to LDS in 32KB chunks (async-to-LDS when available).
// ------------------------------------------------------------------
__global__ void k_composite(const float4* __restrict__ rec,
                            const float* __restrict__ colors,
                            const float* __restrict__ sem,
                            int N, int HW, float* __restrict__ out) {
  __shared__ float4 sdat[CHUNK];
  const int i = blockIdx.x * blockDim.x + threadIdx.x;

  const float4 my = rec[(i < N) ? i : 0];
  const int   pi = __float_as_int(my.x);
  const float di = my.y;
  const float oi = my.w;
  float tsum = 0.0f;

  for (int cbase = 0; cbase < N; cbase += CHUNK) {
    const int cnt = min(CHUNK, N - cbase);
#if __has_builtin(__builtin_amdgcn_global_load_async_to_lds_b128) && \
    __has_builtin(__builtin_amdgcn_s_wait_asynccnt)
    for (int t = threadIdx.x; t < cnt; t += blockDim.x) {
      __builtin_amdgcn_global_load_async_to_lds_b128(
          (__attribute__((address_space(1))) v4i*)(rec + cbase + t),
          (__attribute__((address_space(3))) v4i*)(&sdat[t]),
          0, 0);
    }
    __builtin_amdgcn_s_wait_asynccnt(0);
#else
    for (int t = threadIdx.x; t < cnt; t += blockDim.x)
      sdat[t] = rec[cbase + t];
#endif
    __syncthreads();

    for (int j = 0; j < cnt; ++j) {        // uniform j -> LDS broadcast reads
      const float4 s  = sdat[j];
      const int    pj = __float_as_int(s.x);
      const int    jg = cbase + j;
      const bool nearer = (pj == pi) && ((s.y < di) || ((s.y == di) && (jg < i)));
      tsum += nearer ? s.z : 0.0f;
    }
    __syncthreads();
  }

  if (i < N && pi < HW) {
    const float alpha = oi * expf(tsum);
#pragma unroll
    for (int f = 0; f < 3; ++f)
      atomicAdd(&out[f * HW + pi], alpha * colors[i * 3 + f]);
#pragma unroll
    for (int f = 0; f < 32; ++f)
      atomicAdd(&out[(3 + f) * HW + pi], alpha * sem[i * 32 + f]);
  }
}

// ------------------------------------------------------------------
extern "C" void kernel_launch(void* const* d_in, const int* in_sizes, int n_in,
                              void* d_out, int out_size, void* d_ws, size_t ws_size,
                              hipStream_t stream) {
  const float* centers = (const float*)d_in[0];   // (N,3)
  const float* colors  = (const float*)d_in[1];   // (N,3)
  const float* opac    = (const float*)d_in[2];   // (N,1)
  const float* sem     = (const float*)d_in[3];   // (N,32)
  const float* pose    = (const float*)d_in[4];   // (4,4)
  const float* intr    = (const float*)d_in[5];   // (3,3)
  const int*   Hp      = (const int*)d_in[6];
  const int*   Wp      = (const int*)d_in[7];

  const int N  = in_sizes[0] / 3;
  const int HW = out_size / FEAT_DIM;
  float4* rec = (float4*)d_ws;                    // N * 16 bytes

  // 1) zero the output (bandwidth-dominant step)
  k_zero<<<2048, 256, 0, stream>>>((float*)d_out, out_size);

  // 2) WMMA projection: 16 points per wave
  {
    const int waves   = (N + 15) / 16;
    const int threads = waves * 32;
    const int blocks  = (threads + 255) / 256;
    k_project<<<blocks, 256, 0, stream>>>(centers, opac, pose, intr, Hp, Wp, N, rec);
  }

  // 3) compositing + scatter
  {
    const int blocks = (N + 255) / 256;
    k_composite<<<blocks, 256, 0, stream>>>(rec, colors, sem, N, HW, (float*)d_out);
  }
}